// Top1Gate_2216203125407
// MI455X (gfx1250) — compile-verified
//
#include <hip/hip_runtime.h>
#include <hip/hip_bf16.h>
#include <math.h>

typedef float v2f __attribute__((ext_vector_type(2)));
typedef float v8f __attribute__((ext_vector_type(8)));

#define T_TOK 8192
#define N_EXP 64
#define DIM   4096
#define CAP   128

// ---------------------------------------------------------------------------
// 1) logits = x (8192x4096) * wg^T (4096x64), f32 WMMA 16x16x4.
//    Block = 128 threads = 4 waves; block handles 16 tokens, wave w handles
//    experts [16w, 16w+16). A-frag: lanes 0-15 rows M=0..15 K={0,1},
//    lanes 16-31 K={2,3}. B-frag (K x N): lanes hold N=lane&15, K offset same.
// ---------------------------------------------------------------------------
__global__ __launch_bounds__(128) void logits_gemm(const float* __restrict__ x,
                                                   const float* __restrict__ wg,
                                                   float* __restrict__ logits) {
    const int lane  = threadIdx.x & 31;
    const int wv    = threadIdx.x >> 5;
    const int mbase = blockIdx.x << 4;
    const int koff  = (lane >> 4) << 1;          // 0 or 2
    const int mr    = mbase + (lane & 15);       // token row for A
    const int nr    = (wv << 4) + (lane & 15);   // expert row for B (= wg row)

    const float* __restrict__ arow = x  + (size_t)mr * DIM + koff;
    const float* __restrict__ brow = wg + (size_t)nr * DIM + koff;

    v8f acc = {0.f, 0.f, 0.f, 0.f, 0.f, 0.f, 0.f, 0.f};
#pragma unroll 4
    for (int k = 0; k < DIM; k += 4) {
        v2f a = *(const v2f*)(arow + k);
        v2f b = *(const v2f*)(brow + k);
        acc = __builtin_amdgcn_wmma_f32_16x16x4_f32(false, a, false, b,
                                                    (short)0, acc, false, false);
    }
    // C layout: VGPR r -> M = r (lanes 0-15) or r+8 (lanes 16-31), N = lane&15
    const int mrow = (lane >> 4) << 3;
    const int ncol = (wv << 4) + (lane & 15);
#pragma unroll
    for (int r = 0; r < 8; ++r)
        logits[(size_t)(mbase + mrow + r) * N_EXP + ncol] = acc[r];
}

// ---------------------------------------------------------------------------
// 2) Per-token softmax over 64 experts; one wave per token (2 experts/lane).
//    argmax tie-break = lowest index (matches jnp.argmax). gate_s = 1/sumexp.
// ---------------------------------------------------------------------------
__global__ __launch_bounds__(256) void softmax_gate(const float* __restrict__ logits,
                                                    float* __restrict__ gates,
                                                    int* __restrict__ expert_idx,
                                                    float* __restrict__ gate_s) {
    const int lane = threadIdx.x & 31;
    const int wv   = threadIdx.x >> 5;
    const int t    = blockIdx.x * 8 + wv;

    const float l0 = logits[(size_t)t * N_EXP + lane];
    const float l1 = logits[(size_t)t * N_EXP + lane + 32];

    float v; int i;
    if (l1 > l0) { v = l1; i = lane + 32; } else { v = l0; i = lane; }
#pragma unroll
    for (int off = 16; off > 0; off >>= 1) {
        float ov = __shfl_xor(v, off, 32);
        int   oi = __shfl_xor(i, off, 32);
        if (ov > v || (ov == v && oi < i)) { v = ov; i = oi; }
    }
    const float e0 = expf(l0 - v);
    const float e1 = expf(l1 - v);
    float s = e0 + e1;
#pragma unroll
    for (int off = 16; off > 0; off >>= 1) s += __shfl_xor(s, off, 32);

    const float inv = 1.0f / s;
    gates[(size_t)t * N_EXP + lane]      = e0 * inv;
    gates[(size_t)t * N_EXP + lane + 32] = e1 * inv;
    if (lane == 0) { expert_idx[t] = i; gate_s[t] = inv; }
}

// ---------------------------------------------------------------------------
// 3) locations1: ordered per-expert cumsum over tokens. One block per expert,
//    ballot-based wave prefix + LDS scan of wave totals. Deterministic.
// ---------------------------------------------------------------------------
__global__ __launch_bounds__(256) void scan_locations(const int* __restrict__ expert_idx,
                                                      int* __restrict__ loc,
                                                      int* __restrict__ counts) {
    const int e    = blockIdx.x;
    const int tid  = threadIdx.x;
    const int lane = tid & 31;
    const int wv   = tid >> 5;
    __shared__ int wsum[8];
    __shared__ int woff[8];
    __shared__ int running;
    if (tid == 0) running = 0;
    __syncthreads();

    for (int chunk = 0; chunk < T_TOK; chunk += 256) {
        const int t = chunk + tid;
        const bool flag = (expert_idx[t] == e);
        const unsigned mask = __builtin_amdgcn_ballot_w32(flag);
        const int prefix = __popc(mask & ((1u << lane) - 1u));
        if (lane == 0) wsum[wv] = __popc(mask);
        __syncthreads();
        if (tid == 0) {
            int acc = running;
#pragma unroll
            for (int w = 0; w < 8; ++w) { woff[w] = acc; acc += wsum[w]; }
            running = acc;
        }
        __syncthreads();
        if (flag) loc[t] = woff[wv] + prefix;
        __syncthreads();
    }
    if (tid == 0) counts[e] = running;
}

// ---------------------------------------------------------------------------
// 4) me_sum[e] = sum_t gates[t][e], fixed-order tree reduction (deterministic)
// ---------------------------------------------------------------------------
__global__ __launch_bounds__(256) void me_reduce(const float* __restrict__ gates,
                                                 float* __restrict__ me_sum) {
    const int e = blockIdx.x;
    const int tid = threadIdx.x;
    float s = 0.0f;
    for (int t = tid; t < T_TOK; t += 256) s += gates[(size_t)t * N_EXP + e];
    __shared__ float red[256];
    red[tid] = s;
    __syncthreads();
    for (int off = 128; off > 0; off >>= 1) {
        if (tid < off) red[tid] += red[tid + off];
        __syncthreads();
    }
    if (tid == 0) me_sum[e] = red[0];
}

// ---------------------------------------------------------------------------
// 5) Zero the entire output (bandwidth-critical: ~536 MB), b128 stores.
// ---------------------------------------------------------------------------
__global__ __launch_bounds__(256) void fill_zero(float4* __restrict__ out4, size_t n4,
                                                 float* __restrict__ out, size_t n) {
    size_t i = (size_t)blockIdx.x * blockDim.x + threadIdx.x;
    const size_t stride = (size_t)gridDim.x * blockDim.x;
    const float4 z = make_float4(0.f, 0.f, 0.f, 0.f);
    for (; i < n4; i += stride) out4[i] = z;
    if (blockIdx.x == 0 && threadIdx.x == 0)
        for (size_t j = n4 * 4; j < n; ++j) out[j] = 0.0f;
}

// ---------------------------------------------------------------------------
// 6) l_aux = E * sum_e (me[e]/T) * (counts[e]/T)
// ---------------------------------------------------------------------------
__global__ __launch_bounds__(64) void laux_kernel(const float* __restrict__ me_sum,
                                                  const int* __restrict__ counts,
                                                  float* __restrict__ out) {
    const int e = threadIdx.x;
    const float invT = 1.0f / (float)T_TOK;
    __shared__ float red[64];
    red[e] = (me_sum[e] * invT) * ((float)counts[e] * invT);
    __syncthreads();
    for (int off = 32; off > 0; off >>= 1) {
        if (e < off) red[e] += red[e + off];
        __syncthreads();
    }
    if (e == 0) out[0] = red[0] * (float)N_EXP;
}

// ---------------------------------------------------------------------------
// 7) Sparse scatter: one nonzero per surviving token in combine + dispatch.
// ---------------------------------------------------------------------------
__global__ __launch_bounds__(256) void scatter_kernel(const int* __restrict__ expert_idx,
                                                      const int* __restrict__ loc,
                                                      const float* __restrict__ gate_s,
                                                      float* __restrict__ out,
                                                      size_t plane) {
    const int t = blockIdx.x * blockDim.x + threadIdx.x;
    if (t >= T_TOK) return;
    const int l = loc[t];
    if (l < CAP) {
        const int e = expert_idx[t];
        const float g = gate_s[t];
        const size_t base = 1 + (size_t)t * (N_EXP * CAP) + (size_t)e * CAP + (size_t)l;
        out[base] = g;                                  // combine1_sec
        out[base + plane] = (g != 0.0f) ? 1.0f : 0.0f;  // dispatch_mask
    }
}

extern "C" void kernel_launch(void* const* d_in, const int* in_sizes, int n_in,
                              void* d_out, int out_size, void* d_ws, size_t ws_size,
                              hipStream_t stream) {
    const float* x  = (const float*)d_in[0];
    const float* wg = (const float*)d_in[1];
    float* out = (float*)d_out;

    const size_t plane = (size_t)T_TOK * N_EXP * CAP;   // 67,108,864

    // 2 x 2MB scratch matrices staged in the tail of d_out (zeroed later by fill)
    float* logits = out + (size_t)out_size - 2 * (size_t)T_TOK * N_EXP;
    float* gates  = logits + (size_t)T_TOK * N_EXP;

    // Small per-token arrays in workspace (~100 KB)
    int*   expert_idx = (int*)d_ws;
    float* gate_s     = (float*)(expert_idx + T_TOK);
    int*   loc        = (int*)(gate_s + T_TOK);
    int*   counts     = loc + T_TOK;
    float* me_sum     = (float*)(counts + N_EXP);

    logits_gemm   <<<T_TOK / 16, 128, 0, stream>>>(x, wg, logits);
    softmax_gate  <<<T_TOK / 8,  256, 0, stream>>>(logits, gates, expert_idx, gate_s);
    scan_locations<<<N_EXP,      256, 0, stream>>>(expert_idx, loc, counts);
    me_reduce     <<<N_EXP,      256, 0, stream>>>(gates, me_sum);
    fill_zero     <<<8192,       256, 0, stream>>>((float4*)out, (size_t)out_size / 4,
                                                   out, (size_t)out_size);
    laux_kernel   <<<1,          64,  0, stream>>>(me_sum, counts, out);
    scatter_kernel<<<(T_TOK + 255) / 256, 256, 0, stream>>>(expert_idx, loc, gate_s,
                                                            out, plane);
}